// CtaPostAttnMixer_51247549776563
// MI455X (gfx1250) — compile-verified
//
#include <hip/hip_runtime.h>
#include <hip/hip_bf16.h>
#include <stdint.h>

// ---------------------------------------------------------------------------
// 4-step fixed-boundary 1D heat diffusion along the sequence dim, fused into
// a single HBM pass via a time-skewed register pipeline, with the input tile
// streamed into LDS by the CDNA5 Tensor Data Mover (double-buffered).
// Grid is split over (l-tiles x channel-halves x batch) = 512 workgroups so
// multiple independent TDM streams run per WGP.
// ---------------------------------------------------------------------------

typedef float f32x4 __attribute__((ext_vector_type(4)));
typedef unsigned int u32x4 __attribute__((ext_vector_type(4)));
typedef int i32x4 __attribute__((ext_vector_type(4)));
typedef int i32x8 __attribute__((ext_vector_type(8)));

#define SEQ_L 8192
#define DIM_D 1024
#define DHALF 512          // channels handled per block
#define BATCH 4
#define LTILE 128          // output rows per block
#define CH    8            // rows per TDM chunk
#define NCH   ((LTILE + 8) / CH)   // 17 chunks (128 outputs + 8 halo rows)
#define ALPHA 0.1f

__device__ __forceinline__ f32x4 stencil_or_pass(f32x4 m, f32x4 c, f32x4 p, int gi) {
    // One diffusion step: c + alpha*(p - 2c + m); rows 0 and L-1 pass through.
    f32x4 lap = p - 2.0f * c + m;
    f32x4 s = c + ALPHA * lap;
    bool interior = (gi > 0) & (gi < SEQ_L - 1);
    return interior ? s : c;
}

// Issue one TDM DMA: an n-row x 512-col f32 tile (row stride 1024) -> LDS.
__device__ __forceinline__ void tdm_load_rows(const float* gsrc, uint32_t lds_byte_off,
                                              int rows_remaining, int n) {
    uint64_t ga = (uint64_t)(uintptr_t)gsrc;
    u32x4 g0;
    g0[0] = 1u;                                  // count=1 (valid descriptor)
    g0[1] = lds_byte_off;                        // lds_addr
    g0[2] = (uint32_t)ga;                        // global_addr[31:0]
    g0[3] = ((uint32_t)(ga >> 32) & 0x01FFFFFFu) // global_addr[56:32]
            | 0x80000000u;                       // type=2 ("image")
    i32x8 g1;
    g1[0] = (int)(2u << 16);                     // wg_mask=0, data_size=2 (4B)
    g1[1] = (int)((uint32_t)DHALF << 16);        // tensor_dim0[15:0]=512
    g1[2] = (int)(((uint32_t)rows_remaining & 0xFFFFu) << 16);   // dim0 hi=0 | tensor_dim1 lo
    g1[3] = (int)((((uint32_t)rows_remaining >> 16) & 0xFFFFu)   // tensor_dim1 hi
                  | ((uint32_t)DHALF << 16));                    // tile_dim0=512
    g1[4] = (int)((uint32_t)n & 0xFFFFu);        // tile_dim1=n rows, tile_dim2=0
    g1[5] = (int)DIM_D;                          // tensor_dim0_stride=1024 (full row)
    g1[6] = 0;
    g1[7] = 0;
    i32x4 g2 = {0, 0, 0, 0};
    i32x4 g3 = {0, 0, 0, 0};
#if defined(__clang_major__) && (__clang_major__ >= 23)
    i32x8 g4 = {0, 0, 0, 0, 0, 0, 0, 0};
    __builtin_amdgcn_tensor_load_to_lds(g0, g1, g2, g3, g4, 0);
#else
    __builtin_amdgcn_tensor_load_to_lds(g0, g1, g2, g3, 0);
#endif
}

__global__ __launch_bounds__(128) void heat4_kernel(const float* __restrict__ x,
                                                    float* __restrict__ y) {
    __shared__ float smem[2][CH][DHALF];         // 32 KB double buffer

    const int b  = blockIdx.z;
    const int dh = blockIdx.y;                   // which 512-channel half
    const int T0 = blockIdx.x * LTILE;           // first output row of this tile
    const float* xb = x + (size_t)b * SEQ_L * DIM_D + (size_t)dh * DHALF;
    float*       yb = y + (size_t)b * SEQ_L * DIM_D + (size_t)dh * DHALF;
    const int d0 = threadIdx.x * 4;              // this thread's 4 channels (in half)
    const bool issuer = (threadIdx.x < 32);      // wave 0 drives the TDM

    auto issue_chunk = [&](int c) {
        int R0 = T0 - 4 + c * CH;                // first (logical) row of chunk
        int slotoff = (R0 < 0) ? -R0 : 0;        // negative halo of tile 0
        int gr = R0 + slotoff;                   // first real row loaded
        int n  = CH - slotoff;                   // rows to transfer
        uint32_t lds_off = (uint32_t)(uintptr_t)(&smem[c & 1][slotoff][0]);
        tdm_load_rows(xb + (size_t)gr * DIM_D, lds_off, SEQ_L - gr, n);
    };

    // Time-skew pipeline state: two trailing values of y0..y3 per channel.
    f32x4 s0a = {0,0,0,0}, s0b = {0,0,0,0};
    f32x4 s1a = {0,0,0,0}, s1b = {0,0,0,0};
    f32x4 s2a = {0,0,0,0}, s2b = {0,0,0,0};
    f32x4 s3a = {0,0,0,0}, s3b = {0,0,0,0};

    if (issuer) issue_chunk(0);

    const int jbase = T0 - 4;
    for (int c = 0; c < NCH; ++c) {
        if (issuer) {
            if (c + 1 < NCH) {
                issue_chunk(c + 1);                      // prefetch next chunk
                __builtin_amdgcn_s_wait_tensorcnt(1);    // chunk c is complete
            } else {
                __builtin_amdgcn_s_wait_tensorcnt(0);
            }
        }
        __syncthreads();                                 // chunk c visible to all

#pragma unroll
        for (int r = 0; r < CH; ++r) {
            const int j = jbase + c * CH + r;            // row just loaded: y0[j]
            f32x4 cur = *reinterpret_cast<const f32x4*>(&smem[c & 1][r][d0]);
            f32x4 v1 = stencil_or_pass(s0b, s0a, cur, j - 1);  // y1[j-1]
            f32x4 v2 = stencil_or_pass(s1b, s1a, v1,  j - 2);  // y2[j-2]
            f32x4 v3 = stencil_or_pass(s2b, s2a, v2,  j - 3);  // y3[j-3]
            f32x4 v4 = stencil_or_pass(s3b, s3a, v3,  j - 4);  // y4[j-4]
            if (c > 0 || r >= 4) {                       // j-4 in [T0-4, T0+LTILE)
                f32x4* dst = reinterpret_cast<f32x4*>(&yb[(size_t)(j - 4) * DIM_D + d0]);
                __builtin_nontemporal_store(v4, dst);    // streamed, never re-read
            }
            s0b = s0a; s0a = cur;
            s1b = s1a; s1a = v1;
            s2b = s2a; s2a = v2;
            s3b = s3a; s3a = v3;
        }
        __syncthreads();                                 // buffer safe to overwrite
    }
}

extern "C" void kernel_launch(void* const* d_in, const int* in_sizes, int n_in,
                              void* d_out, int out_size, void* d_ws, size_t ws_size,
                              hipStream_t stream) {
    const float* x = (const float*)d_in[0];
    float* y = (float*)d_out;
    dim3 grid(SEQ_L / LTILE, DIM_D / DHALF, BATCH);   // 64 x 2 x 4 = 512 blocks
    dim3 block(128);                                  // 128 threads = 512 channels
    heat4_kernel<<<grid, block, 0, stream>>>(x, y);
}